// Criterion_20426864460201
// MI455X (gfx1250) — compile-verified
//
#include <hip/hip_runtime.h>
#include <hip/hip_bf16.h>
#include <math.h>

typedef float v2f __attribute__((ext_vector_type(2)));
typedef float v8f __attribute__((ext_vector_type(8)));

// Sum a value across all 32 lanes of a wave using V_WMMA_F32_16X16X4_F32.
// A-layout (32-bit A 16x4): VGPR0 lanes0-15 -> A[m][0]=p_m, lanes16-31 -> A[m][2]=p_{m+16};
// VGPR1 (K=1,3) zeroed. With B = all-ones, D[m][n] = p_m + p_{m+16}.
// D-layout: VGPR j, lanes0-15 hold M=j, lanes16-31 hold M=j+8. Summing the 8 D VGPRs
// gives half-sums per 16-lane half; one xor-16 shuffle completes the reduction.
__device__ __forceinline__ float wave_reduce_sum(float p) {
    v2f a; a.x = p;    a.y = 0.0f;
    v2f b; b.x = 1.0f; b.y = 1.0f;
    v8f c = {};
    c = __builtin_amdgcn_wmma_f32_16x16x4_f32(
            /*neg_a=*/false, a, /*neg_b=*/false, b,
            /*c_mod=*/(short)0, c, /*reuse_a=*/false, /*reuse_b=*/false);
    float h = 0.0f;
#pragma unroll
    for (int j = 0; j < 8; ++j) h += c[j];
    h += __shfl_xor(h, 16, 32);
    return h;  // total of all 32 lane values, broadcast to every lane
}

// Block-wide sum (256 threads = 8 wave32). All threads call convergently.
__device__ __forceinline__ float block_reduce_sum(float v, float* lds) {
    float w = wave_reduce_sum(v);
    const int wid  = threadIdx.x >> 5;
    const int lane = threadIdx.x & 31;
    __syncthreads();                 // guard LDS reuse across successive reductions
    if (lane == 0) lds[wid] = w;
    __syncthreads();
    float s = 0.0f;
#pragma unroll
    for (int i = 0; i < 8; ++i) s += lds[i];
    return s;
}

// Kernel A: streaming MSE partial sums. n4 = total_elements / 4.
__global__ void __launch_bounds__(256)
mse_partial_kernel(const float4* __restrict__ y, const float4* __restrict__ m,
                   float* __restrict__ partials, int n4) {
    float acc = 0.0f;
    const int stride = gridDim.x * blockDim.x;
    for (int i = blockIdx.x * blockDim.x + threadIdx.x; i < n4; i += stride) {
        float4 a = y[i];
        float4 b = m[i];
        float d0 = a.x - b.x, d1 = a.y - b.y, d2 = a.z - b.z, d3 = a.w - b.w;
        acc = fmaf(d0, d0, acc);
        acc = fmaf(d1, d1, acc);
        acc = fmaf(d2, d2, acc);
        acc = fmaf(d3, d3, acc);
    }
    __shared__ float lds[8];
    float s = block_reduce_sum(acc, lds);
    if (threadIdx.x == 0) partials[blockIdx.x] = s;
}

// Kernel B: reduce partials + last-row Gaussian log-prob terms, write scalar loss.
__global__ void __launch_bounds__(256)
finalize_kernel(const float* __restrict__ partials, int nblk,
                const float* __restrict__ mu, const float* __restrict__ sigma,
                const float* __restrict__ ty, int bs, int nt,
                float* __restrict__ out) {
    // 1) sum of squared diffs from kernel A
    float smse = 0.0f;
    for (int i = threadIdx.x; i < nblk; i += blockDim.x) smse += partials[i];

    // 2) last-row terms: Q = sum((x-m)^2 / v), L = sum(log v)
    const size_t off = (size_t)(bs - 1) * (size_t)nt;
    const float4* x4 = (const float4*)(ty + off);
    const float4* m4 = (const float4*)(mu + off);
    const float4* v4 = (const float4*)(sigma + off);
    float q = 0.0f, l = 0.0f;
    const int n4 = nt >> 2;
    for (int i = threadIdx.x; i < n4; i += blockDim.x) {
        float4 x = x4[i], m = m4[i], v = v4[i];
        float d;
        d = x.x - m.x; q += d * d / v.x; l += logf(v.x);
        d = x.y - m.y; q += d * d / v.y; l += logf(v.y);
        d = x.z - m.z; q += d * d / v.z; l += logf(v.z);
        d = x.w - m.w; q += d * d / v.w; l += logf(v.w);
    }

    __shared__ float lds[8];
    float S = block_reduce_sum(smse, lds);
    float Q = block_reduce_sum(q, lds);
    float L = block_reduce_sum(l, lds);

    if (threadIdx.x == 0) {
        const float LOG2PI = 1.8378770664093453f;   // log(2*pi)
        float fbs = (float)bs, fnt = (float)nt;
        float log_prob = -0.5f * (Q + L + fnt * LOG2PI);
        float loss1 = -log_prob / fnt;              // per-dim NLL of last sample
        float mse = S / (fbs * fnt);
        float loss = loss1 / fbs + 0.5f * mse;      // LAM = 0.5
        out[0] = loss / fbs;
    }
}

extern "C" void kernel_launch(void* const* d_in, const int* in_sizes, int n_in,
                              void* d_out, int out_size, void* d_ws, size_t ws_size,
                              hipStream_t stream) {
    const float* mu       = (const float*)d_in[0];
    const float* sigma    = (const float*)d_in[1];
    const float* target_y = (const float*)d_in[2];
    const float* moran_y  = (const float*)d_in[3];
    const float* moran_mu = (const float*)d_in[4];
    float* out = (float*)d_out;
    float* partials = (float*)d_ws;

    const int n  = in_sizes[3];      // bs * nt = 8388608
    const int nt = 4096;             // per reference shapes
    const int bs = n / nt;           // 2048
    const int n4 = n >> 2;

    const int blocks = 1024;         // spreads across WGPs; 32 elems/thread
    mse_partial_kernel<<<blocks, 256, 0, stream>>>(
        (const float4*)moran_y, (const float4*)moran_mu, partials, n4);
    finalize_kernel<<<1, 256, 0, stream>>>(
        partials, blocks, mu, sigma, target_y, bs, nt, out);
}